// SGCN_27307402068688
// MI455X (gfx1250) — compile-verified
//
#include <hip/hip_runtime.h>
#include <hip/hip_bf16.h>

typedef __attribute__((ext_vector_type(2))) float v2f;
typedef __attribute__((ext_vector_type(8))) float v8f;

#define F_IN   128
#define DM     256
#define NCLS   10
#define NGRAPH 128

// ---------------------------------------------------------------- degree / dinv
__global__ void k_deg_init(float* __restrict__ deg, int n) {
    int i = blockIdx.x * blockDim.x + threadIdx.x;
    if (i < n) deg[i] = 1.0f;   // self-loop contribution
}

__global__ void k_deg_count(const int* __restrict__ dst, float* __restrict__ deg, int e) {
    int i = blockIdx.x * blockDim.x + threadIdx.x;
    if (i < e) atomicAdd(&deg[dst[i]], 1.0f);
}

__global__ void k_rsqrt_inplace(float* __restrict__ deg, int n) {
    int i = blockIdx.x * blockDim.x + threadIdx.x;
    if (i < n) deg[i] = rsqrtf(deg[i]);   // deg >= 1 always (self-loop)
}

// ---------------------------------------------------------------- propagation
// out[v, :] = dinv[v]^2 * in[v, :]   (self-loop term; also initializes out)
__global__ void k_selfloop(const float* __restrict__ in, float* __restrict__ out,
                           const float* __restrict__ dinv, int n, int F) {
    long long t = (long long)blockIdx.x * blockDim.x + threadIdx.x;
    int chunks = F >> 2;
    long long total = (long long)n * chunks;
    if (t >= total) return;
    int v = (int)(t / chunks);
    int c = (int)(t % chunks);
    float w = dinv[v] * dinv[v];
    const float4 a = *(const float4*)(in + (size_t)v * F + c * 4);
    float4 r; r.x = w * a.x; r.y = w * a.y; r.z = w * a.z; r.w = w * a.w;
    *(float4*)(out + (size_t)v * F + c * 4) = r;
}

// out[dst, :] += dinv[src]*dinv[dst] * in[src, :]  per edge (float atomics).
// 32 consecutive threads share one edge: src/dst reads broadcast, feature row
// reads fully coalesced (float4 per lane).
__global__ void k_edge_scatter(const float* __restrict__ in, float* __restrict__ out,
                               const int* __restrict__ src, const int* __restrict__ dst,
                               const float* __restrict__ dinv, int e, int F) {
    long long t = (long long)blockIdx.x * blockDim.x + threadIdx.x;
    int chunks = F >> 2;
    long long total = (long long)e * chunks;
    if (t >= total) return;
    int ei = (int)(t / chunks);
    int c  = (int)(t % chunks);
    int s = src[ei], d = dst[ei];
    float w = dinv[s] * dinv[d];
    const float4 a = *(const float4*)(in + (size_t)s * F + c * 4);
    float* o = out + (size_t)d * F + c * 4;
    atomicAdd(o + 0, w * a.x);
    atomicAdd(o + 1, w * a.y);
    atomicAdd(o + 2, w * a.z);
    atomicAdd(o + 3, w * a.w);
}

__global__ void k_bias_relu(float* __restrict__ h, const float* __restrict__ b,
                            int n, int F) {
    long long t = (long long)blockIdx.x * blockDim.x + threadIdx.x;
    int chunks = F >> 2;
    long long total = (long long)n * chunks;
    if (t >= total) return;
    int v = (int)(t / chunks);
    int c = (int)(t % chunks);
    float4 a = *(const float4*)(h + (size_t)v * F + c * 4);
    const float4 bb = *(const float4*)(b + c * 4);
    a.x = fmaxf(a.x + bb.x, 0.f);
    a.y = fmaxf(a.y + bb.y, 0.f);
    a.z = fmaxf(a.z + bb.z, 0.f);
    a.w = fmaxf(a.w + bb.w, 0.f);
    *(float4*)(h + (size_t)v * F + c * 4) = a;
}

// ---------------------------------------------------------------- WMMA fp32 GEMM
// C[M,Nc] = (optionally relu)( A[M,K] @ W[K,Nc] + bias )
// One wave32 computes a 16x64 output strip: one shared A fragment feeds 4
// independent V_WMMA_F32_16X16X4_F32 accumulators per K-step (amortizes A
// traffic 4x; independent D regs avoid the WMMA->WMMA RAW hazard NOPs).
// Per-lane layouts (ISA 7.12.2): A 16x4 -> 2 VGPRs: lanes 0-15 rows M=lane
// K={k,k+1}, lanes 16-31 same rows K={k+2,k+3}; B 4x16 mirrors with N striped
// across lanes. C: VGPR r -> row m0+r (lanes 0-15) / m0+8+r (lanes 16-31).
__global__ void k_wmma_gemm(const float* __restrict__ A, const float* __restrict__ W,
                            const float* __restrict__ bias, float* __restrict__ C,
                            int M, int K, int Nc, int addBias, int doRelu) {
    int wave = threadIdx.x >> 5;
    int lane = threadIdx.x & 31;
    int tile = blockIdx.x * (blockDim.x >> 5) + wave;
    int strips = Nc >> 6;                 // 64-wide column strips
    int tilesM = M >> 4;                  // M multiple of 16, Nc multiple of 64
    if (tile >= tilesM * strips) return;  // wave-uniform exit, EXEC stays full
    int m0 = (tile / strips) << 4;
    int n0 = (tile % strips) << 6;
    int lh  = lane >> 4;                  // 0: K pair {k,k+1}; 1: {k+2,k+3}
    int l15 = lane & 15;

    const float* arow = A + (size_t)(m0 + l15) * K + lh * 2;
    const float* wrow = W + (size_t)(lh * 2) * Nc + n0 + l15;

    v8f acc0 = {}, acc1 = {}, acc2 = {}, acc3 = {};
    for (int k = 0; k < K; k += 4) {
        v2f a;
        a.x = arow[k];
        a.y = arow[k + 1];
        const float* w0 = wrow + (size_t)k * Nc;
        const float* w1 = wrow + (size_t)(k + 1) * Nc;
        v2f b0, b1, b2, b3;
        b0.x = w0[0];  b0.y = w1[0];
        b1.x = w0[16]; b1.y = w1[16];
        b2.x = w0[32]; b2.y = w1[32];
        b3.x = w0[48]; b3.y = w1[48];
        acc0 = __builtin_amdgcn_wmma_f32_16x16x4_f32(false, a, false, b0, (short)0, acc0, false, false);
        acc1 = __builtin_amdgcn_wmma_f32_16x16x4_f32(false, a, false, b1, (short)0, acc1, false, false);
        acc2 = __builtin_amdgcn_wmma_f32_16x16x4_f32(false, a, false, b2, (short)0, acc2, false, false);
        acc3 = __builtin_amdgcn_wmma_f32_16x16x4_f32(false, a, false, b3, (short)0, acc3, false, false);
    }

    v8f accs[4] = {acc0, acc1, acc2, acc3};
#pragma unroll
    for (int j = 0; j < 4; ++j) {
        int col = n0 + j * 16 + l15;
        float bv = addBias ? bias[col] : 0.0f;
#pragma unroll
        for (int r = 0; r < 8; ++r) {
            int row = m0 + r + lh * 8;
            float v = accs[j][r] + bv;
            if (doRelu) v = fmaxf(v, 0.0f);
            C[(size_t)row * Nc + col] = v;
        }
    }
}

// ---------------------------------------------------------------- pooling + head
__global__ void k_zero(float* __restrict__ p, int n) {
    int i = blockIdx.x * blockDim.x + threadIdx.x;
    if (i < n) p[i] = 0.0f;
}

__global__ void k_pool(const float* __restrict__ h, const int* __restrict__ batch,
                       float* __restrict__ g, int n, int F) {
    long long t = (long long)blockIdx.x * blockDim.x + threadIdx.x;
    int chunks = F >> 2;
    long long total = (long long)n * chunks;
    if (t >= total) return;
    int v = (int)(t / chunks);
    int c = (int)(t % chunks);
    int b = batch[v];
    const float4 a = *(const float4*)(h + (size_t)v * F + c * 4);
    float* o = g + (size_t)b * F + c * 4;
    atomicAdd(o + 0, a.x);
    atomicAdd(o + 1, a.y);
    atomicAdd(o + 2, a.z);
    atomicAdd(o + 3, a.w);
}

__global__ void k_fc3(const float* __restrict__ g2, const float* __restrict__ w,
                      const float* __restrict__ b, float* __restrict__ out, int K) {
    int t = blockIdx.x * blockDim.x + threadIdx.x;
    if (t >= NGRAPH * NCLS) return;
    int gi = t / NCLS, c = t % NCLS;
    float acc = b[c];
    for (int k = 0; k < K; ++k) acc += g2[gi * K + k] * w[k * NCLS + c];
    out[t] = acc;
}

// ---------------------------------------------------------------- launcher
static inline int cdiv_ll(long long a, int b) { return (int)((a + b - 1) / b); }

static void launch_gemm(const float* A, const float* W, const float* bias, float* C,
                        int M, int K, int Nc, int addBias, int doRelu,
                        hipStream_t stream) {
    int tiles = (M >> 4) * (Nc >> 6);         // 16x64 strips per wave
    int blocks = (tiles + 3) / 4;             // 4 waves (128 threads) per block
    k_wmma_gemm<<<blocks, 128, 0, stream>>>(A, W, bias, C, M, K, Nc, addBias, doRelu);
}

static void launch_propagate(const float* in, float* out, const int* src, const int* dst,
                             const float* dinv, int n, int e, int F, hipStream_t stream) {
    long long nt = (long long)n * (F >> 2);
    k_selfloop<<<cdiv_ll(nt, 256), 256, 0, stream>>>(in, out, dinv, n, F);
    long long et = (long long)e * (F >> 2);
    k_edge_scatter<<<cdiv_ll(et, 256), 256, 0, stream>>>(in, out, src, dst, dinv, e, F);
}

extern "C" void kernel_launch(void* const* d_in, const int* in_sizes, int n_in,
                              void* d_out, int out_size, void* d_ws, size_t ws_size,
                              hipStream_t stream) {
    const float* x      = (const float*)d_in[0];
    const int*   ei     = (const int*)d_in[1];
    const int*   batch  = (const int*)d_in[2];
    const float* sg_w   = (const float*)d_in[3];
    const float* sg_b   = (const float*)d_in[4];
    const float* gcn_w  = (const float*)d_in[5];
    const float* gcn_b  = (const float*)d_in[6];
    const float* fc1_w  = (const float*)d_in[7];
    const float* fc1_b  = (const float*)d_in[8];
    const float* fc2_w  = (const float*)d_in[9];
    const float* fc2_b  = (const float*)d_in[10];
    const float* fc3_w  = (const float*)d_in[11];
    const float* fc3_b  = (const float*)d_in[12];

    const int N = in_sizes[0] / F_IN;
    const int E = in_sizes[1] / 2;
    const int* src = ei;
    const int* dst = ei + E;

    // workspace layout (floats)
    float* ws   = (float*)d_ws;
    float* bufA = ws;                              // [N, 256]
    float* bufB = bufA + (size_t)N * DM;           // [N, 256]
    float* dinv = bufB + (size_t)N * DM;           // [N]
    float* g0   = dinv + N;                        // [128, 256]
    float* g1   = g0 + NGRAPH * DM;                // [128, 256]
    float* g2   = g1 + NGRAPH * DM;                // [128, 128]

    // 1) dinv = rsqrt(in_deg + 1)
    k_deg_init<<<cdiv_ll(N, 256), 256, 0, stream>>>(dinv, N);
    k_deg_count<<<cdiv_ll(E, 256), 256, 0, stream>>>(dst, dinv, E);
    k_rsqrt_inplace<<<cdiv_ll(N, 256), 256, 0, stream>>>(dinv, N);

    // 2) SGConv: S^2 x, then relu(. @ sg_w + sg_b)   (F=128 propagations)
    launch_propagate(x,    bufA, src, dst, dinv, N, E, F_IN, stream);
    launch_propagate(bufA, bufB, src, dst, dinv, N, E, F_IN, stream);
    launch_gemm(bufB, sg_w, sg_b, bufA, N, F_IN, DM, 1, 1, stream);   // h in bufA [N,256]

    // 3) three GCN layers: relu(S (h @ Wi) + bi)
    for (int i = 0; i < 3; ++i) {
        launch_gemm(bufA, gcn_w + (size_t)i * DM * DM, nullptr, bufB,
                    N, DM, DM, 0, 0, stream);                          // bufB = h @ Wi
        launch_propagate(bufB, bufA, src, dst, dinv, N, E, DM, stream);// bufA = S bufB
        long long nt = (long long)N * (DM >> 2);
        k_bias_relu<<<cdiv_ll(nt, 256), 256, 0, stream>>>(bufA, gcn_b + (size_t)i * DM, N, DM);
    }

    // 4) global_add_pool -> g0 [128, 256]
    k_zero<<<cdiv_ll(NGRAPH * DM, 256), 256, 0, stream>>>(g0, NGRAPH * DM);
    {
        long long nt = (long long)N * (DM >> 2);
        k_pool<<<cdiv_ll(nt, 256), 256, 0, stream>>>(bufA, batch, g0, N, DM);
    }

    // 5) MLP head
    launch_gemm(g0, fc1_w, fc1_b, g1, NGRAPH, DM, DM, 1, 1, stream);       // [128,256]
    launch_gemm(g1, fc2_w, fc2_b, g2, NGRAPH, DM, DM / 2, 1, 1, stream);   // [128,128]
    k_fc3<<<cdiv_ll(NGRAPH * NCLS, 128), 128, 0, stream>>>(g2, fc3_w, fc3_b,
                                                           (float*)d_out, DM / 2);
}